// KWinners2d_34170759807260
// MI455X (gfx1250) — compile-verified
//
#include <hip/hip_runtime.h>

#define CHN    256
#define IMG_H  56
#define IMG_W  56
#define HW     (IMG_H * IMG_W)     // 3136
#define NBATCH 32
#define KSEL   26                  // k winners per pixel (compile-time)
#define BLOCK  256

// ---------------- gfx1250 async-to-LDS path (guarded) ----------------------
#if __has_builtin(__builtin_amdgcn_global_load_async_to_lds_b32)
#define USE_ASYNC_LDS 1
#else
#define USE_ASYNC_LDS 0
#endif

#if __has_builtin(__builtin_amdgcn_s_wait_asynccnt)
#define WAIT_ASYNC(n) __builtin_amdgcn_s_wait_asynccnt(n)
#else
#define WAIT_ASYNC(n) asm volatile("s_wait_asynccnt %0" ::"n"(n) : "memory")
#endif

typedef __attribute__((address_space(1))) int* as1_i32_ptr;
typedef __attribute__((address_space(3))) int* as3_i32_ptr;

__device__ __forceinline__ as1_i32_ptr to_glb(const void* p) {
    return (as1_i32_ptr)(uintptr_t)p;           // generic -> global AS
}
__device__ __forceinline__ as3_i32_ptr to_lds(void* p) {
    return (as3_i32_ptr)(uint32_t)(uintptr_t)p; // low 32 bits = LDS offset
}

#if USE_ASYNC_LDS
#define CH_CHUNK   32
#define NCHUNK     (CHN / CH_CHUNK)   // 8
#define LDS_STRIDE 36                 // dword stride per lane column (bank spread)
#endif
// ---------------------------------------------------------------------------

// Insert v into descending top-K list: L'[i] = med3(L[i-1], L[i], v).
// One v_med3_num_f32 per slot, all slots independent -> 27 VALU/element.
__device__ __forceinline__ void topk_insert(float (&list)[KSEL], float v) {
#pragma unroll
    for (int i = KSEL - 1; i >= 1; --i)
        list[i] = __builtin_amdgcn_fmed3f(list[i - 1], list[i], v);
    list[0] = fmaxf(list[0], v);
}

__global__ __launch_bounds__(BLOCK) void kwinners2d_kernel(
    const float* __restrict__ x,
    const float* __restrict__ duty,     // 256 per-channel duty cycles
    const float* __restrict__ boost,    // scalar boost strength
    float* __restrict__ out)
{
    __shared__ __align__(16) float s_scale[CHN];
#if USE_ASYNC_LDS
    __shared__ __align__(16) float s_buf[2][BLOCK][LDS_STRIDE]; // 72KB double buffer
#endif

    const int tid = threadIdx.x;

    // Per-channel scale = exp(-boost * duty[c]); wave-uniform broadcast later.
    s_scale[tid] = expf(-boost[0] * duty[tid]);

    // Pixel mapping: consecutive lanes -> consecutive w -> coalesced 128B lines.
    const long p  = (long)blockIdx.x * BLOCK + tid;   // < NBATCH*HW
    const int  b  = (int)(p / HW);
    const int  hw = (int)(p % HW);
    const size_t base = (size_t)b * ((size_t)CHN * HW) + (size_t)hw;
    const float* __restrict__ xp = x + base;
    float*       __restrict__ op = out + base;

    float list[KSEL];
#pragma unroll
    for (int i = 0; i < KSEL; ++i) list[i] = -__builtin_inff();

#if USE_ASYNC_LDS
    // Kick off chunk 0 before the barrier (does not depend on s_scale).
    float* lcol0 = &s_buf[0][tid][0];
#pragma unroll
    for (int cc = 0; cc < CH_CHUNK; ++cc)
        __builtin_amdgcn_global_load_async_to_lds_b32(
            to_glb(xp + (size_t)cc * HW), to_lds(lcol0 + cc), 0, 0);
#endif

    __syncthreads();   // s_scale ready

#if USE_ASYNC_LDS
    // ---- Pass 1: double-buffered async staging, compute from LDS ----
    int pb = 0;
    for (int ck = 0; ck < NCHUNK; ++ck) {
        if (ck + 1 < NCHUNK) {
            float* lcol = &s_buf[pb ^ 1][tid][0];
            const int c0 = (ck + 1) * CH_CHUNK;
#pragma unroll
            for (int cc = 0; cc < CH_CHUNK; ++cc)
                __builtin_amdgcn_global_load_async_to_lds_b32(
                    to_glb(xp + (size_t)(c0 + cc) * HW), to_lds(lcol + cc), 0, 0);
            WAIT_ASYNC(CH_CHUNK);   // only next chunk may remain outstanding
        } else {
            WAIT_ASYNC(0);
        }
        asm volatile("" ::: "memory");  // don't hoist LDS reads above the wait

        const float4* lrow = (const float4*)&s_buf[pb][tid][0];
        const float4* srow = (const float4*)&s_scale[ck * CH_CHUNK];
#pragma unroll
        for (int q = 0; q < CH_CHUNK / 4; ++q) {
            const float4 xv = lrow[q];   // ds_load_b128
            const float4 sc = srow[q];
            topk_insert(list, xv.x * sc.x);
            topk_insert(list, xv.y * sc.y);
            topk_insert(list, xv.z * sc.z);
            topk_insert(list, xv.w * sc.w);
        }
        pb ^= 1;
    }
#else
    // ---- Pass 1 (fallback): direct global streaming with prefetch ----
#pragma unroll 4
    for (int c = 0; c < CHN; ++c) {
        if ((c & 3) == 0 && c + 16 < CHN)
            __builtin_prefetch(xp + (size_t)(c + 16) * HW, 0, 1); // global_prefetch_b8
        const float xv = xp[(size_t)c * HW];
        topk_insert(list, xv * s_scale[c]);
    }
#endif

    const float thr = list[KSEL - 1];   // k-th largest boosted value

    // ---- Pass 2: re-read x (hits WGP$/L2), apply mask (ties kept, ">=") ----
#pragma unroll 4
    for (int c = 0; c < CHN; ++c) {
        const float xv = xp[(size_t)c * HW];
        const float bv = xv * s_scale[c];
        op[(size_t)c * HW] = (bv >= thr) ? xv : 0.0f;
    }
}

extern "C" void kernel_launch(void* const* d_in, const int* in_sizes, int n_in,
                              void* d_out, int out_size, void* d_ws, size_t ws_size,
                              hipStream_t stream)
{
    // setup_inputs order: x (f32), duty_cycle (f32), k (int, ==26), boost (f32)
    const float* x     = (const float*)d_in[0];
    const float* duty  = (const float*)d_in[1];
    const float* boost = (const float*)d_in[3];
    float* out = (float*)d_out;

    const int pixels = NBATCH * HW;     // 100,352
    const int blocks = pixels / BLOCK;  // exactly 392

    kwinners2d_kernel<<<dim3(blocks), dim3(BLOCK), 0, stream>>>(x, duty, boost, out);
}